// HistogramLoss_76665166233919
// MI455X (gfx1250) — compile-verified
//
#include <hip/hip_runtime.h>
#include <math.h>

// HistogramLoss fused kernel for gfx1250 (MI455X).
// S = F F^T via V_WMMA_F32_16X16X4_F32 (exact fp32). Operand panels are staged
// into LDS by the Tensor Data Mover (TENSOR_LOAD_TO_LDS, one DMA per 64xKC
// panel, hardware row padding -> conflict-free ds_load_b64 fragment reads),
// double-buffered on TENSORcnt. Histogram in LDS ds_add_f32, global-atomic
// reduction, cumsum epilogue.

#define TSIZE 151
#define WS_POS 0
#define WS_NEG TSIZE
#define WS_CNT (2 * TSIZE)   // [0]=pos count, [1]=neg count
#define WS_TOT (2 * TSIZE + 2)

#define KC 128                 // K floats per staged chunk (= TDM tile_dim0)
#define STRIDE (KC + 4)        // TDM pads 4 dwords per 128 -> stride%64(dw)==4
#define PANEL (64 * STRIDE)    // floats per 64-row panel stage

typedef __attribute__((ext_vector_type(2))) float    v2f;
typedef __attribute__((ext_vector_type(8))) float    v8f;
typedef __attribute__((ext_vector_type(4))) unsigned v4u;
typedef __attribute__((ext_vector_type(8))) unsigned v8u;

// One TDM descriptor-driven DMA: 64 rows x KC floats from global (row-major,
// row stride d) into LDS at lds_byte, padding +4 dwords after each 128 dwords.
__device__ __forceinline__ void tdm_load_panel(unsigned lds_byte, const float* gptr,
                                               unsigned grow0, unsigned kb,
                                               unsigned du, unsigned nu) {
    const unsigned long long ga = (unsigned long long)(const void*)gptr
                                + ((unsigned long long)grow0 * du + kb) * 4ull;
    v4u g0;
    g0.x = 1u;                                            // count=1, no gather
    g0.y = lds_byte;                                      // lds_addr (bytes)
    g0.z = (unsigned)(ga & 0xFFFFFFFFull);                // global_addr[31:0]
    g0.w = (unsigned)((ga >> 32) & 0x1FFFFFFull)          // global_addr[56:32]
         | (2u << 30);                                    // type=2 ("image")
    v8u g1;
    g1.s0 = (2u << 16)            // data_size=2 -> 4 bytes
          | (1u << 20)            // pad_enable
          | (6u << 22)            // pad_interval=6 -> every 128 dwords
          | (3u << 25);           // pad_amount=3 -> 4 dwords
    g1.s1 = (du & 0xFFFFu) << 16;                         // tensor_dim0[15:0]
    g1.s2 = (du >> 16) | ((nu & 0xFFFFu) << 16);          // tdim0[31:16]|tdim1[15:0]
    g1.s3 = (nu >> 16) | ((unsigned)KC << 16);            // tdim1[31:16]|tile_dim0
    g1.s4 = 64u;                                          // tile_dim1=64, tile_dim2=0
    g1.s5 = du;                                           // tensor_dim0_stride[31:0]
    g1.s6 = 0u;                                           // stride0 hi | stride1 lo
    g1.s7 = 0u;
    const v4u gz = (v4u){0u, 0u, 0u, 0u};                 // groups 2/3: inert (2D)
    asm volatile("tensor_load_to_lds %0, %1, %2, %3"
                 :: "s"(g0), "s"(g1), "s"(gz), "s"(gz)
                 : "memory");
}

__global__ void hist_zero_kernel(float* __restrict__ ws) {
    int t = blockIdx.x * blockDim.x + threadIdx.x;
    if (t < WS_TOT) ws[t] = 0.0f;
}

__global__ __launch_bounds__(256) void hist_tiles_kernel(
        const float* __restrict__ F, const int* __restrict__ cls,
        float* __restrict__ ws, int n, int d) {
    __shared__ float smem[2 * 2 * PANEL];   // [buf][panel A/B][64*STRIDE]
    __shared__ float lh[WS_TOT];

    const int tid = threadIdx.x;
    for (int t = tid; t < WS_TOT; t += 256) lh[t] = 0.0f;
    __syncthreads();

    const int Bj = blockIdx.x;   // column super-tile
    const int Bi = blockIdx.y;   // row super-tile
    float cpos = 0.0f, cneg = 0.0f;

    if (Bi <= Bj) {
        const int wave = tid >> 5;
        const int lane = tid & 31;
        const int rr   = lane & 15;
        const int half = lane >> 4;
        const int kh   = half * 2;
        const unsigned smem_base = (unsigned)(size_t)(void*)smem;
        const unsigned rowA0 = (unsigned)(Bi * 64);
        const unsigned rowB0 = (unsigned)(Bj * 64);
        const unsigned du = (unsigned)d;
        const unsigned nu = (unsigned)n;

        // Each of 8 waves owns sub-tiles s=wave and s=wave+8 of the 4x4 grid.
        int i0[2], j0[2];
        bool ok[2];
        v8f acc[2];
        #pragma unroll
        for (int t = 0; t < 2; ++t) {
            const int s  = wave + t * 8;
            const int ti = s >> 2, tj = s & 3;
            i0[t] = Bi * 64 + ti * 16;
            j0[t] = Bj * 64 + tj * 16;
            ok[t] = (j0[t] >= i0[t]);
            acc[t] = (v8f){0.f, 0.f, 0.f, 0.f, 0.f, 0.f, 0.f, 0.f};
        }

        const int nc = d / KC;

        // Wave 0 drives the TDM: one DMA per panel per chunk.
        auto issue = [&](int c, int buf) {
            const unsigned kb   = (unsigned)(c * KC);
            const unsigned bufB = smem_base + (unsigned)buf * (2u * PANEL * 4u);
            tdm_load_panel(bufB,               F, rowA0, kb, du, nu);
            tdm_load_panel(bufB + PANEL * 4u,  F, rowB0, kb, du, nu);
        };

        if (wave == 0) issue(0, 0);
        for (int c = 0; c < nc; ++c) {
            if (wave == 0) __builtin_amdgcn_s_wait_tensorcnt(0);  // chunk c landed
            __syncthreads();   // publish LDS; also orders prior reads vs rewrite
            if (c + 1 < nc && wave == 0) issue(c + 1, (c + 1) & 1);

            const float* bufA = smem + (size_t)((c & 1) * 2 * PANEL);
            const float* bufB = bufA + PANEL;

            #pragma unroll
            for (int t = 0; t < 2; ++t) {
                if (!ok[t]) continue;                 // wave-uniform
                const int s  = wave + t * 8;
                const int ti = s >> 2, tj = s & 3;
                // A fragment (16x4) and B fragment (4x16 of F^T) share the
                // same per-lane pattern: row=lane&15, K offset 2*(lane>>4).
                const float* arow = bufA + (ti * 16 + rr) * STRIDE + kh;
                const float* brow = bufB + (tj * 16 + rr) * STRIDE + kh;
                v8f a_ = acc[t];
                #pragma unroll 8
                for (int k = 0; k < KC; k += 4) {
                    v2f a = *(const v2f*)(arow + k);
                    v2f b = *(const v2f*)(brow + k);
                    a_ = __builtin_amdgcn_wmma_f32_16x16x4_f32(
                            false, a, false, b, (short)0, a_, false, false);
                }
                acc[t] = a_;
            }
        }

        // Histogram epilogue. C/D layout: VGPR v, lanes 0-15 -> M=v, N=lane;
        // lanes 16-31 -> M=v+8, N=lane-16.
        const float invstep = (float)(TSIZE - 1) * 0.5f;  // 1/step = 75
        #pragma unroll
        for (int t = 0; t < 2; ++t) {
            if (!ok[t]) continue;
            const int j  = j0[t] + rr;
            const int cj = cls[j];
            #pragma unroll
            for (int v = 0; v < 8; ++v) {
                const int i = i0[t] + v + half * 8;
                if (j > i) {
                    const bool  eq = (cls[i] == cj);
                    const float sv = acc[t][v];
                    float x    = (sv + 1.0f) * invstep;
                    float fidx = floorf(x);
                    fidx = fminf(fmaxf(fidx, 0.0f), (float)(TSIZE - 1));
                    const int   idx    = (int)fidx;
                    const float frac   = x - fidx;
                    const float w_lo   = 1.0f - frac;
                    const int   idx_hi = (idx + 1 < TSIZE) ? idx + 1 : TSIZE - 1;
                    const float w_hi   = (idx + 1 < TSIZE) ? frac : 0.0f;
                    float* h = eq ? (lh + WS_POS) : (lh + WS_NEG);
                    atomicAdd(h + idx, w_lo);        // ds_add_f32
                    atomicAdd(h + idx_hi, w_hi);
                    if (eq) cpos += 1.0f; else cneg += 1.0f;
                }
            }
        }
        if (cpos != 0.0f) atomicAdd(&lh[WS_CNT + 0], cpos);
        if (cneg != 0.0f) atomicAdd(&lh[WS_CNT + 1], cneg);
    }

    __syncthreads();
    for (int t = tid; t < WS_TOT; t += 256) {
        const float v = lh[t];
        if (v != 0.0f) atomicAdd(&ws[t], v);   // global_atomic_add_f32
    }
}

__global__ void hist_final_kernel(const float* __restrict__ ws,
                                  float* __restrict__ out) {
    if (threadIdx.x == 0 && blockIdx.x == 0) {
        const float ps = ws[WS_CNT + 0];
        const float ns = ws[WS_CNT + 1];
        float cdf = 0.0f, loss = 0.0f;
        for (int t = 0; t < TSIZE; ++t) {
            cdf  += ws[WS_POS + t] / ps;
            loss += (ws[WS_NEG + t] / ns) * cdf;
        }
        out[0] = loss;
    }
}

extern "C" void kernel_launch(void* const* d_in, const int* in_sizes, int n_in,
                              void* d_out, int out_size, void* d_ws, size_t ws_size,
                              hipStream_t stream) {
    const float* F   = (const float*)d_in[0];   // [N, D] fp32, L2-normalized rows
    const int*   cls = (const int*)d_in[1];     // [N] int32
    float*       out = (float*)d_out;           // scalar
    float*       ws  = (float*)d_ws;            // >= WS_TOT floats

    const int n = in_sizes[1];
    const int d = in_sizes[0] / n;              // 4096, 512
    const int Ns = n / 64;                      // 64 super-tiles per dim

    hipLaunchKernelGGL(hist_zero_kernel, dim3(1), dim3(512), 0, stream, ws);
    hipLaunchKernelGGL(hist_tiles_kernel, dim3(Ns, Ns), dim3(256), 0, stream,
                       F, cls, ws, n, d);
    hipLaunchKernelGGL(hist_final_kernel, dim3(1), dim3(32), 0, stream, ws, out);
}